// Maximum_Path_Generater_4277787427516
// MI455X (gfx1250) — compile-verified
//
#include <hip/hip_runtime.h>
#include <hip/hip_bf16.h>
#include <stdint.h>

// Problem constants (fixed by the reference setup_inputs()).
constexpr int B = 32;
constexpr int T = 512;   // tokens
constexpr int F = 2048;  // mel frames
constexpr float NEGC = -1e7f;

// ---------------------------------------------------------------------------
// Helpers: gfx1250 async global->LDS copy + ASYNCcnt waits (inline CDNA5 asm)
// ---------------------------------------------------------------------------

// Flat->LDS: per CDNA5 ISA, an LDS-aperture flat address maps to LDS via its
// low 32 bits, so truncation yields the LDS byte offset for async-load VDST.
__device__ __forceinline__ unsigned lds_off(const void* p) {
  return (unsigned)(unsigned long long)p;
}

__device__ __forceinline__ void async_load_b128_to_lds(unsigned lds_addr,
                                                       const void* gaddr) {
  asm volatile("global_load_async_to_lds_b128 %0, %1, off"
               :
               : "v"(lds_addr), "v"((unsigned long long)gaddr)
               : "memory");
}

__device__ __forceinline__ void wait_async_le4() {
  asm volatile("s_wait_asynccnt 0x4" ::: "memory");
}
__device__ __forceinline__ void wait_async_le0() {
  asm volatile("s_wait_asynccnt 0x0" ::: "memory");
}

// ---------------------------------------------------------------------------
// Kernel 1: fused mask-multiply + transpose  [B,T,F] -> [B,F,T]
// (written into d_out, reused as scratch; DP columns become contiguous 2KB)
// ---------------------------------------------------------------------------
__global__ __launch_bounds__(256)
void transpose_mask_kernel(const float* __restrict__ log_p,
                           const float* __restrict__ mask,
                           float* __restrict__ lpT) {
  __shared__ float tile[32][33];
  const int b  = blockIdx.z;
  const int f0 = blockIdx.x * 32;
  const int t0 = blockIdx.y * 32;
  const int tx = threadIdx.x;   // 0..31
  const int ty = threadIdx.y;   // 0..7
  const size_t base = (size_t)b * T * F;

#pragma unroll
  for (int k = 0; k < 4; ++k) {
    const int t = t0 + ty + k * 8;
    const size_t idx = base + (size_t)t * F + (f0 + tx);
    tile[ty + k * 8][tx] = log_p[idx] * mask[idx];
  }
  __syncthreads();
#pragma unroll
  for (int k = 0; k < 4; ++k) {
    const int ff = f0 + ty + k * 8;
    const size_t oidx = (size_t)b * F * T + (size_t)ff * T + (t0 + tx);
    lpT[oidx] = tile[tx][ty + k * 8];
  }
}

// ---------------------------------------------------------------------------
// Kernel 2: per-batch Viterbi DP (one wave32 per batch) + in-LDS backtrace.
// Dynamic LDS layout:
//   [0,      2048)  : column buffer 0 (512 floats)
//   [2048,   4096)  : column buffer 1 (512 floats)
//   [4096, 135168)  : move-decision bits, u16[F*32]  (128 KB, CDNA5 big LDS)
// ---------------------------------------------------------------------------
__device__ __forceinline__ void load_col_swz(const float4* colq, int lane,
                                             float lp[16]) {
#pragma unroll
  for (int j = 0; j < 4; ++j) {
    const int g  = lane * 4 + j;          // logical 16B granule of the column
    const int ph = g ^ ((g >> 4) & 3);    // XOR swizzle -> conflict-free b128
    const float4 v = colq[ph];
    lp[j * 4 + 0] = v.x;
    lp[j * 4 + 1] = v.y;
    lp[j * 4 + 2] = v.z;
    lp[j * 4 + 3] = v.w;
  }
}

__global__ __launch_bounds__(32)
void mas_dp_kernel(const float* __restrict__ lpT,
                   const float* __restrict__ mask,
                   int* __restrict__ path_tok) {
  extern __shared__ __align__(16) char smem[];
  float* colbuf0 = (float*)smem;
  float* colbuf1 = (float*)(smem + 2048);
  unsigned short* bits = (unsigned short*)(smem + 4096);

  const int b = blockIdx.x;
  const int lane = threadIdx.x;
  const float* lpT_b = lpT + (size_t)b * F * T;

  // ---- lengths from mask (tok_len = sum_t mask[b,t,0]; feat = sum_f mask[b,0,f])
  float tl = 0.f, fl = 0.f;
  {
    const size_t mbase = (size_t)b * T * F;
#pragma unroll
    for (int i = 0; i < 16; ++i)
      tl += mask[mbase + (size_t)(lane * 16 + i) * F];
    for (int j = lane; j < F; j += 32) fl += mask[mbase + j];
#pragma unroll
    for (int off = 16; off >= 1; off >>= 1) {
      tl += __shfl_xor(tl, off, 32);
      fl += __shfl_xor(fl, off, 32);
    }
  }
  const int tok_len  = (int)(tl + 0.5f);
  const int feat_len = (int)(fl + 0.5f);

  const unsigned cb0 = lds_off(colbuf0);
  const unsigned cb1 = lds_off(colbuf1);

  auto issue_col = [&](int f, int buf) {
    const unsigned base = buf ? cb1 : cb0;
#pragma unroll
    for (int k = 0; k < 4; ++k) {
      const int w = k * 32 + lane;                      // granule to fetch
      const unsigned dst = base + (unsigned)((w ^ ((w >> 4) & 3)) << 4);
      const float* s = lpT_b + (size_t)f * T + (size_t)w * 4;
      async_load_b128_to_lds(dst, s);
    }
  };

  float x[16];  // x[t, f-1] for t = lane*16 + i

  // prologue: prefetch columns 0 and 1
  issue_col(0, 0);
  issue_col(1, 1);
  wait_async_le4();  // column 0 landed (column 1 still in flight)
  asm volatile("" ::: "memory");

  // ---- column 0 (closed form): x[t,0] = (t==0 ? 0 : NEG) + lp[t,0]
  {
    float lp[16];
    load_col_swz((const float4*)colbuf0, lane, lp);
#pragma unroll
    for (int i = 0; i < 16; ++i) {
      const int t = lane * 16 + i;
      x[i] = ((t == 0) ? 0.f : NEGC) + lp[i];
    }
    bits[lane] = (lane == 0) ? (unsigned short)1u : (unsigned short)0u;
  }

  // ---- main DP over columns, async double-buffered
  for (int f = 1; f < F; ++f) {
    if (f + 1 < F) {
      issue_col(f + 1, (f + 1) & 1);
      wait_async_le4();  // oldest 4 async ops (column f) complete, in order
    } else {
      wait_async_le0();
    }
    asm volatile("" ::: "memory");

    float lp[16];
    load_col_swz((const float4*)((f & 1) ? colbuf1 : colbuf0), lane, lp);

    float boundary = __shfl_up(x[15], 1, 32);  // x[lane*16 - 1, f-1]
    if (lane == 0) boundary = NEGC;            // also encodes t==0 prev_q = NEG

    unsigned mb = 0;
    float nx[16];
    if (f < T) {  // diagonal constraint t==f only possible while f < T
#pragma unroll
      for (int i = 0; i < 16; ++i) {
        const int t = lane * 16 + i;
        const float xm1 = (i == 0) ? boundary : x[i - 1];
        const bool diag = (t == f);
        const float cur = diag ? NEGC : x[i];
        mb |= ((unsigned)(diag | (x[i] < xm1))) << i;
        nx[i] = fmaxf(cur, xm1) + lp[i];
      }
    } else {
#pragma unroll
      for (int i = 0; i < 16; ++i) {
        const float xm1 = (i == 0) ? boundary : x[i - 1];
        mb |= ((unsigned)(x[i] < xm1)) << i;
        nx[i] = fmaxf(x[i], xm1) + lp[i];
      }
    }
#pragma unroll
    for (int i = 0; i < 16; ++i) x[i] = nx[i];
    bits[f * 32 + lane] = (unsigned short)mb;
  }

  // ---- backtrace from (tok_len-1, F-1), entirely from LDS decision bits
  if (lane == 0) {
    int t = tok_len - 1;
    int* pt_b = path_tok + b * F;
    for (int f = F - 1; f >= 0; --f) {
      const bool active = (f < feat_len);
      pt_b[f] = active ? t : -1;
      if (active) {
        const unsigned w = bits[f * 32 + (t >> 4)];
        if ((w >> (t & 15)) & 1) t = (t > 0) ? (t - 1) : 0;
      }
    }
  }
}

// ---------------------------------------------------------------------------
// Kernel 3: expand path_tok[b,f] -> one-hot path[b,t,f] (coalesced f4 stores)
// ---------------------------------------------------------------------------
__global__ __launch_bounds__(256)
void expand_path_kernel(const int* __restrict__ path_tok,
                        float* __restrict__ out) {
  const size_t gid = (size_t)blockIdx.x * 256 + threadIdx.x;  // over B*T*(F/4)
  constexpr int F4 = F / 4;
  const int f4 = (int)(gid % F4);
  const int bt = (int)(gid / F4);
  const int t = bt & (T - 1);
  const int b = bt >> 9;  // T == 512
  const int4 p = ((const int4*)path_tok)[b * F4 + f4];
  float4 o;
  o.x = (p.x == t) ? 1.f : 0.f;
  o.y = (p.y == t) ? 1.f : 0.f;
  o.z = (p.z == t) ? 1.f : 0.f;
  o.w = (p.w == t) ? 1.f : 0.f;
  ((float4*)out)[gid] = o;
}

// ---------------------------------------------------------------------------
extern "C" void kernel_launch(void* const* d_in, const int* in_sizes, int n_in,
                              void* d_out, int out_size, void* d_ws,
                              size_t ws_size, hipStream_t stream) {
  (void)in_sizes; (void)n_in; (void)out_size; (void)ws_size;
  const float* log_p = (const float*)d_in[0];
  const float* mask  = (const float*)d_in[1];
  float* out = (float*)d_out;
  int* path_tok = (int*)d_ws;  // B*F ints = 256 KB of workspace

  // Stage 1: fused mask*log_p transpose into d_out (reused as [B,F,T] scratch;
  // it is fully overwritten by stage 3).
  float* lpT = out;
  dim3 gT(F / 32, T / 32, B), bT(32, 8);
  transpose_mask_kernel<<<gT, bT, 0, stream>>>(log_p, mask, lpT);

  // Stage 2: per-batch DP + backtrace. Dynamic LDS: 2 column buffers (4 KB)
  // + 128 KB of move-decision bits (CDNA5 320KB-per-WGP LDS makes this legal).
  const size_t dp_smem = 2 * T * sizeof(float) + (size_t)F * 32 * sizeof(unsigned short);
  mas_dp_kernel<<<B, 32, dp_smem, stream>>>(lpT, mask, path_tok);

  // Stage 3: expand to one-hot [B,T,F] output.
  const int n4 = (B * T * (F / 4)) / 256;  // 32768 blocks
  expand_path_kernel<<<n4, 256, 0, stream>>>(path_tok, out);
}